// NeuralNetwork_5763846111598
// MI455X (gfx1250) — compile-verified
//
#include <hip/hip_runtime.h>
#include <hip/hip_bf16.h>
#include <math.h>

// ---------------------------------------------------------------------------
// B=1 recurrent net -> chain of bandwidth-bound GEMVs on gfx1250 (MI455X).
// GEMV core: V_WMMA_F32_16X16X4_F32. A wave owns 16 rows of W; the x-chunk is
// broadcast across B's 16 N-columns so every D column carries the same 16
// row-dots. Ping-pong 64-K chunks keep 16-32 global_load_b64 (4-8 KB/wave)
// outstanding with partial loadcnt waits; serpentine row-tile order across
// the 3 recurrent steps recovers ~192 MB/step of W_h2h from the 192 MB L2.
// Split-K(8) partials in exclusive slots keep the FP reduction deterministic.
// ---------------------------------------------------------------------------

typedef __attribute__((ext_vector_type(2))) float v2f;
typedef __attribute__((ext_vector_type(8))) float v8f;

#define KSPLIT 8
#define ROWS_PER_WAVE 16
#define WAVES_PER_BLOCK 8
#define ROWS_PER_BLOCK (ROWS_PER_WAVE * WAVES_PER_BLOCK)  // 128
#define CHUNK 64                                          // K-values per stage
#define NV2F (CHUNK / 4)                                  // 16 v2f per stage
#define LEAKY_SLOPE 0.01f

__device__ __forceinline__ v8f wmma_f32_k4(v2f a, v2f b, v8f c)
{
    return __builtin_amdgcn_wmma_f32_16x16x4_f32(
        false, a, false, b, (short)0, c, false, false);
}

__device__ __forceinline__ void load_chunk(v2f (&buf)[NV2F],
                                           const float* __restrict__ wp, int kb)
{
#pragma unroll
    for (int u = 0; u < NV2F; ++u)
        buf[u] = *(const v2f*)(wp + kb + 4 * u);
}

__device__ __forceinline__ void consume_chunk(const v2f (&buf)[NV2F],
                                              const float* __restrict__ xp, int kb,
                                              v8f& acc0, v8f& acc1)
{
#pragma unroll
    for (int u = 0; u < NV2F; ++u) {
        v2f b = *(const v2f*)(xp + kb + 4 * u);    // broadcast LDS read
        if (u & 1) acc1 = wmma_f32_k4(buf[u], b, acc1);
        else       acc0 = wmma_f32_k4(buf[u], b, acc0);
    }
}

// part[kseg][row] (+)= sum over K-segment of W[row, k] * x[k]
// grid = (nrows/128, KSPLIT), block = 256, dyn LDS = (K/KSPLIT)*4 bytes.
__global__ __launch_bounds__(256) void gemv_part_kernel(
    const float* __restrict__ W, const float* __restrict__ x,
    float* __restrict__ part, int K, int nrows, int accumulate, int reverse)
{
    const int seg  = K / KSPLIT;          // multiple of 2*CHUNK for all our K
    const int kseg = blockIdx.y;
    const int k0   = kseg * seg;

    // serpentine row-tile order: lets consecutive W_h2h passes hit L2
    const int rblk = reverse ? (gridDim.x - 1 - blockIdx.x) : blockIdx.x;

    extern __shared__ float xs[];         // this block's x segment
    for (int i = threadIdx.x; i < seg; i += blockDim.x)
        xs[i] = x[k0 + i];
    __syncthreads();

    const int wave = threadIdx.x >> 5;
    const int lane = threadIdx.x & 31;
    const int m    = lane & 15;           // row within the 16-row tile
    const int kh2  = (lane >> 4) << 1;    // 0 for lanes 0-15, 2 for 16-31
    const int row_base = rblk * ROWS_PER_BLOCK + wave * ROWS_PER_WAVE;

    const float* __restrict__ wp  = W + (size_t)(row_base + m) * (size_t)K + k0 + kh2;
    const float* __restrict__ xp  = xs + kh2;
    // prefetch pointer: lane-halves skewed by 32 floats so the two halves
    // cover different 128B lines of each row
    const float* __restrict__ pfp = wp + ((lane >> 4) << 5);

    v8f acc0 = {};
    v8f acc1 = {};

    v2f bufA[NV2F], bufB[NV2F];
    load_chunk(bufA, wp, 0);              // 16 loads in flight
    load_chunk(bufB, wp, CHUNK);          // 32 loads in flight

    int kb = 0;
    for (; kb + 4 * CHUNK <= seg; kb += 2 * CHUNK) {
        __builtin_prefetch(pfp + kb + 1024, 0, 0);     // global_prefetch_b8
        __builtin_prefetch(pfp + kb + 1088, 0, 0);
        consume_chunk(bufA, xp, kb, acc0, acc1);
        load_chunk(bufA, wp, kb + 2 * CHUNK);
        consume_chunk(bufB, xp, kb + CHUNK, acc0, acc1);
        load_chunk(bufB, wp, kb + 3 * CHUNK);
    }
    // last two chunks are already in registers (kb == seg - 2*CHUNK)
    consume_chunk(bufA, xp, kb, acc0, acc1);
    consume_chunk(bufB, xp, kb + CHUNK, acc0, acc1);

    // Every D column is identical; lane 0 holds M=0..7, lane 16 holds M=8..15
    // (f32 C/D layout). Those two lanes publish the 16 row results.
    if ((lane & 15) == 0) {
        const int rb = row_base + ((lane >> 4) << 3);
        float* __restrict__ dst = part + (size_t)kseg * (size_t)nrows + rb;
        if (accumulate) {
#pragma unroll
            for (int v = 0; v < 8; ++v) dst[v] += acc0[v] + acc1[v];
        } else {
#pragma unroll
            for (int v = 0; v < 8; ++v) dst[v] = acc0[v] + acc1[v];
        }
    }
}

__device__ __forceinline__ float apply_act(float x, int id)
{
    switch (id) {
    case 0:  return x;
    case 1:  return (x >= 0.0f) ? x : LEAKY_SLOPE * x;       // leaky_relu
    case 2:  return fmaxf(x, 0.0f);                          // relu
    case 3:  return 1.0f / (1.0f + __expf(-x));              // sigmoid
    default: return tanhf(x);                                // tanh
    }
}

// base[r] = sum_k part[k][r]   (fixed-order deterministic reduction)
__global__ __launch_bounds__(256) void combine_kernel(
    const float* __restrict__ part, int nrows, float* __restrict__ base)
{
    const int r = blockIdx.x * blockDim.x + threadIdx.x;
    if (r >= nrows) return;
    float s = 0.0f;
#pragma unroll
    for (int k = 0; k < KSPLIT; ++k) s += part[(size_t)k * nrows + r];
    base[r] = s;
}

// out[r] = act(resp[r] * (base[r] + sum_k part[k][r]) + bias[r], ids[r])
__global__ __launch_bounds__(256) void epilogue_kernel(
    const float* __restrict__ base, const float* __restrict__ part,
    const float* __restrict__ resp, const float* __restrict__ bias,
    const int* __restrict__ ids, float* __restrict__ out, int nrows)
{
    const int r = blockIdx.x * blockDim.x + threadIdx.x;
    if (r >= nrows) return;
    float s = base[r];
#pragma unroll
    for (int k = 0; k < KSPLIT; ++k) s += part[(size_t)k * nrows + r];
    s = resp[r] * s + bias[r];
    out[r] = apply_act(s, ids[r]);
}

extern "C" void kernel_launch(void* const* d_in, const int* in_sizes, int n_in,
                              void* d_out, int out_size, void* d_ws, size_t ws_size,
                              hipStream_t stream)
{
    (void)in_sizes; (void)n_in; (void)out_size; (void)ws_size;

    const float* x     = (const float*)d_in[0];   // (1, NI)
    const float* Wi2h  = (const float*)d_in[1];   // (NH, NI)
    const float* Wh2h  = (const float*)d_in[2];   // (NH, NH)
    const float* Wo2h  = (const float*)d_in[3];   // (NH, NO)
    const float* Wi2o  = (const float*)d_in[4];   // (NO, NI)
    const float* Wh2o  = (const float*)d_in[5];   // (NO, NH)
    const float* Wo2o  = (const float*)d_in[6];   // (NO, NO)
    const float* hresp = (const float*)d_in[7];
    const float* hbias = (const float*)d_in[8];
    const float* oresp = (const float*)d_in[9];
    const float* obias = (const float*)d_in[10];
    const float* h0    = (const float*)d_in[11];  // activs0
    const float* out0  = (const float*)d_in[12];  // outputs0 (loop-invariant)
    const int*   hids  = (const int*)d_in[13];
    const int*   oids  = (const int*)d_in[14];
    float* out = (float*)d_out;

    const int NI = 4096, NH = 8192, NO = 2048;

    // workspace layout (floats)
    float* ws     = (float*)d_ws;
    float* base_h = ws;                            // NH
    float* base_o = base_h + NH;                   // NO
    float* hA     = base_o + NO;                   // NH
    float* hB     = hA + NH;                       // NH
    float* hpart  = hB + NH;                       // KSPLIT * NH
    float* opart  = hpart + (size_t)KSPLIT * NH;   // KSPLIT * NO

    const dim3 blk(256);
    auto gemv = [&](const float* W, const float* v, int K, int nrows,
                    float* part, int accflag, int reverse) {
        dim3 grid(nrows / ROWS_PER_BLOCK, KSPLIT);
        size_t shmem = (size_t)(K / KSPLIT) * sizeof(float);
        gemv_part_kernel<<<grid, blk, shmem, stream>>>(
            W, v, part, K, nrows, accflag, reverse);
    };

    // base_h = Wi2h @ x + Wo2h @ out0   (loop-invariant, streamed once)
    gemv(Wi2h, x,    NI, NH, hpart, 0, 0);
    gemv(Wo2h, out0, NO, NH, hpart, 1, 0);
    combine_kernel<<<NH / 256, blk, 0, stream>>>(hpart, NH, base_h);

    // base_o = Wi2o @ x + Wo2o @ out0
    gemv(Wi2o, x,    NI, NO, opart, 0, 0);
    gemv(Wo2o, out0, NO, NO, opart, 1, 0);
    combine_kernel<<<NO / 256, blk, 0, stream>>>(opart, NO, base_o);

    // 3 recurrent steps: h = act(resp*(base_h + Wh2h @ h) + bias)
    // Serpentine W_h2h traversal (fwd, rev, fwd) -> ~192 MB/step L2 reuse.
    const float* prev = h0;
    float* bufs[2] = { hA, hB };
    for (int s = 0; s < 3; ++s) {
        float* next = bufs[s & 1];
        gemv(Wh2h, prev, NH, NH, hpart, 0, s & 1);
        epilogue_kernel<<<NH / 256, blk, 0, stream>>>(
            base_h, hpart, hresp, hbias, hids, next, NH);
        prev = next;
    }

    // output head: out = act(oresp*(base_o + Wh2o @ h3) + obias)
    gemv(Wh2o, prev, NH, NO, opart, 0, 0);
    epilogue_kernel<<<NO / 256, blk, 0, stream>>>(
        base_o, opart, oresp, obias, oids, out, NO);
}